// exactConeAlignedCosine_70385924046969
// MI455X (gfx1250) — compile-verified
//
#include <hip/hip_runtime.h>
#include <hip/hip_bf16.h>
#include <math.h>
#include <stdint.h>

#define B_SZ 256
#define K_SZ 256
#define D_SZ 512
#define DC   16            // D-chunk per LDS stage (multiple of WMMA K=4)
#define SD   20            // padded LDS row stride in floats (TDM pad: 16+4)
#define NCHUNK (D_SZ / DC) // 32
#define POWER_ITERS 20
#define N_ITERS 200
#define EPS 1e-8f

typedef __attribute__((ext_vector_type(2))) float v2f;
typedef __attribute__((ext_vector_type(8))) float v8f;
typedef __attribute__((ext_vector_type(4))) unsigned int v4u;
typedef __attribute__((ext_vector_type(8))) int v8i;
typedef __attribute__((ext_vector_type(4))) int v4i;

#define G_FLOATS   65536            // 256x256 Gram in LDS
#define BUF_FLOATS (SD * 256)       // 5120 floats = 20480 B per chunk buffer
#define LDS_BYTES  ((G_FLOATS + 2 * BUF_FLOATS) * 4)   // 303104 B

// ---------------------------------------------------------------------------
// Block-wide sum over 1024 threads; `red` is 1024 floats of LDS scratch.
// ---------------------------------------------------------------------------
__device__ __forceinline__ float block_reduce_sum(float* red, float val, int tid) {
    red[tid] = val;
    __syncthreads();
    #pragma unroll
    for (int s = 512; s > 0; s >>= 1) {
        if (tid < s) red[tid] += red[tid + s];
        __syncthreads();
    }
    float r = red[0];
    __syncthreads();
    return r;
}

// ---------------------------------------------------------------------------
// w = G * x for 256x256 symmetric G in LDS. Conflict-free: lanes sweep k.
// ---------------------------------------------------------------------------
__device__ __forceinline__ void matvec256(const float* Gl, const float* x,
                                          float* w, float* red, int tid) {
    const int k    = tid & 255;
    const int part = tid >> 8;
    const float* gj = Gl + (size_t)(part * 64) * 256 + k;
    const float* xv = x + part * 64;
    float s = 0.f;
    #pragma unroll 8
    for (int j = 0; j < 64; ++j) s = fmaf(gj[(size_t)j * 256], xv[j], s);
    red[tid] = s;
    __syncthreads();
    if (tid < 256)
        w[tid] = red[tid] + red[tid + 256] + red[tid + 512] + red[tid + 768];
    __syncthreads();
}

// ---------------------------------------------------------------------------
// Issue a TDM load of ctr[b][:, d0:d0+16] into LDS at lds_byte, with the D#
// pad feature inserting 4 DWORDs per 16-DWORD row -> LDS row stride 20 floats.
// D# bitfields per CDNA5 ISA ch.8 (2D tile: groups 2/3 + aux group zero).
// ---------------------------------------------------------------------------
__device__ __forceinline__ void tdm_load_chunk(const float* ctrb, int d0,
                                               unsigned lds_byte) {
    const unsigned long long ga =
        (unsigned long long)(uintptr_t)ctrb + (unsigned long long)(d0 * 4);

    v4u g0;
    g0.x = 1u;                                         // count=1, user mode
    g0.y = lds_byte;                                   // lds_addr (bytes)
    g0.z = (unsigned)(ga & 0xFFFFFFFFull);             // global_addr[31:0]
    g0.w = (unsigned)((ga >> 32) & 0x01FFFFFFull)      // global_addr[56:32]
         | (2u << 30);                                 // type=2 (image)

    v8i g1;
    g1[0] = (int)((2u << 16)        // data_size = 4 bytes
                | (1u << 20)        // pad_enable
                | (3u << 22)        // pad_interval: 16 DWORDs
                | (3u << 25));      // pad_amount:   4 DWORDs
    g1[1] = (int)(512u << 16);      // tensor_dim0 = 512 (bits 79:48, low 16)
    g1[2] = (int)(256u << 16);      // dim0 hi=0 | tensor_dim1 = 256
    g1[3] = (int)(16u << 16);       // dim1 hi=0 | tile_dim0 = 16
    g1[4] = 256;                    // tile_dim1 = 256, tile_dim2 = 0
    g1[5] = 512;                    // tensor_dim0_stride = 512 (low 32)
    g1[6] = 0;                      // stride0 hi | tensor_dim1_stride (unused)
    g1[7] = 0;

    const v4i gz4 = {};
    const v8i gz8 = {};
    __builtin_amdgcn_tensor_load_to_lds(g0, g1, gz4, gz4, gz8, 0);
}

// ---------------------------------------------------------------------------
// One workgroup per sample. 1024 threads = 32 waves.
// Phase 1: G = ctr ctr^T via V_WMMA_F32_16X16X4_F32, chunks DMA'd by the TDM.
// Phases 2-5: bvec, power iteration, 200 FISTA iterations, cosine loss.
// ---------------------------------------------------------------------------
__global__ void __launch_bounds__(1024, 1)
cone_loss_kernel(const float* __restrict__ pred,
                 const float* __restrict__ ctr,
                 float* __restrict__ ws) {
    extern __shared__ float smem[];
    float* G    = smem;                        // 65536 floats
    float* buf0 = smem + G_FLOATS;             // 5120 floats (padded chunk)
    float* buf1 = smem + G_FLOATS + BUF_FLOATS;
    float* scratch = smem + G_FLOATS;          // reused after phase 1

    const int tid  = threadIdx.x;
    const int b    = blockIdx.x;
    const float* ctrb = ctr + (size_t)b * K_SZ * D_SZ;

    const int wave = tid >> 5;
    const int lane = tid & 31;
    const int ti   = wave >> 1;             // tile row 0..15
    const int tjb  = (wave & 1) * 8;        // first of 8 tile cols
    const int m    = lane & 15;
    const int kb   = (lane >> 4) * 2;       // lanes 16-31 hold K+2,K+3
    const int row_a = ti * 16 + m;

    const unsigned ldsb0 = (unsigned)(G_FLOATS * 4);
    const unsigned ldsb1 = (unsigned)(G_FLOATS * 4 + BUF_FLOATS * 4);

    // ---------------- Phase 1: Gram via FP32 WMMA + TDM double buffer -------
    v8f acc[8];
    {
        const v8f vz = {};
        #pragma unroll
        for (int q = 0; q < 8; ++q) acc[q] = vz;
    }

    if (tid < 32) tdm_load_chunk(ctrb, 0, ldsb0);   // wave 0 issues DMA

    for (int c = 0; c < NCHUNK; ++c) {
        if (tid < 32) {
            if (c + 1 < NCHUNK) {
                tdm_load_chunk(ctrb, (c + 1) * DC, (c & 1) ? ldsb0 : ldsb1);
                __builtin_amdgcn_s_wait_tensorcnt(1);   // chunk c done
            } else {
                __builtin_amdgcn_s_wait_tensorcnt(0);
            }
        }
        if (c + 2 < NCHUNK)
            __builtin_prefetch(ctrb + (size_t)(tid & 255) * D_SZ + (c + 2) * DC, 0, 0);
        __syncthreads();                                // chunk c visible to all

        const float* cur = (c & 1) ? buf1 : buf0;
        #pragma unroll
        for (int kk = 0; kk < DC; kk += 4) {
            const v2f a = *(const v2f*)(cur + row_a * SD + kk + kb);
            #pragma unroll
            for (int q = 0; q < 8; ++q) {
                const int n = (tjb + q) * 16 + m;
                const v2f bb = *(const v2f*)(cur + n * SD + kk + kb);
                acc[q] = __builtin_amdgcn_wmma_f32_16x16x4_f32(
                    false, a, false, bb, (short)0, acc[q], false, false);
            }
        }
        __syncthreads();                 // compute done before buffer reuse
    }

    // Store accumulators to G in LDS (C layout: VGPR r -> rows r / r+8)
    #pragma unroll
    for (int q = 0; q < 8; ++q) {
        const int col = (tjb + q) * 16 + m;
        #pragma unroll
        for (int r = 0; r < 8; ++r) {
            const int row = ti * 16 + r + (lane >> 4) * 8;
            G[row * 256 + col] = acc[q][r];
        }
    }
    __syncthreads();

    // ---------------- Scratch layout (aliases chunk buffers) ----------------
    float* bvec = scratch;          // 256
    float* vv   = scratch + 256;    // 256
    float* w    = scratch + 512;    // 256
    float* lam  = scratch + 768;    // 256
    float* yv   = scratch + 1024;   // 256
    float* red  = scratch + 2048;   // 1024
    float* cpl  = scratch + 3072;   // 512

    // ---------------- Phase 2: cp = -pred, bvec = ctr * cp, ||cp|| ----------
    if (tid < 512) cpl[tid] = -pred[(size_t)b * D_SZ + tid];
    __syncthreads();
    {
        const int k  = tid >> 2;
        const int qq = tid & 3;
        const float* rowp = ctrb + (size_t)k * D_SZ + qq * 128;
        const float* cpp  = cpl + qq * 128;
        float s = 0.f;
        #pragma unroll 8
        for (int i = 0; i < 128; ++i) s = fmaf(rowp[i], cpp[i], s);
        red[tid] = s;
    }
    __syncthreads();
    if (tid < 256)
        bvec[tid] = red[tid * 4] + red[tid * 4 + 1] + red[tid * 4 + 2] + red[tid * 4 + 3];
    __syncthreads();
    const float cv  = (tid < 512) ? cpl[tid] : 0.f;
    const float ncp = sqrtf(block_reduce_sum(red, cv * cv, tid));

    // ---------------- Phase 3: power iteration -> step = 1/L ----------------
    if (tid < 256) vv[tid] = 1.0f;
    __syncthreads();
    for (int it = 0; it < POWER_ITERS; ++it) {
        matvec256(G, vv, w, red, tid);
        const float wt  = (tid < 256) ? w[tid] : 0.f;
        const float nrm = sqrtf(block_reduce_sum(red, wt * wt, tid));
        if (tid < 256) vv[tid] = w[tid] / (nrm + EPS);
        __syncthreads();
    }
    matvec256(G, vv, w, red, tid);
    float step;
    {
        const float wt = (tid < 256) ? w[tid] : 0.f;
        const float L  = sqrtf(block_reduce_sum(red, wt * wt, tid)) * 1.01f + EPS;
        step = 1.0f / L;
    }

    // ---------------- Phase 4: FISTA (200 iterations from LDS) --------------
    if (tid < 256) { lam[tid] = 0.f; yv[tid] = 0.f; }
    float tt = 1.0f;
    __syncthreads();
    for (int it = 0; it < N_ITERS; ++it) {
        matvec256(G, yv, w, red, tid);
        const float tn = 0.5f * (1.0f + sqrtf(1.0f + 4.0f * tt * tt));
        if (tid < 256) {
            const float g  = w[tid] - bvec[tid];
            const float ln = fmaxf(yv[tid] - step * g, 0.f);
            const float yn = ln + ((tt - 1.0f) / tn) * (ln - lam[tid]);
            lam[tid] = ln;
            yv[tid]  = yn;
        }
        tt = tn;
        __syncthreads();
    }

    // ---------------- Phase 5: proj + cosine loss ---------------------------
    float pd = 0.f;
    if (tid < 512) {
        const float* cc = ctrb + tid;
        float s = 0.f;
        #pragma unroll 8
        for (int k2 = 0; k2 < 256; ++k2) s = fmaf(cc[(size_t)k2 * D_SZ], lam[k2], s);
        pd = s;
    }
    __syncthreads();
    const float p2    = block_reduce_sum(red, pd * pd, tid);
    const float dotcp = block_reduce_sum(red, ((tid < 512) ? cpl[tid] : 0.f) * pd, tid);
    if (tid == 0) {
        const float ln  = sqrtf(p2);
        const float num = dotcp / (ln + EPS);
        const float npn = ln / (ln + EPS);
        const float den = fmaxf(ncp, EPS) * fmaxf(npn, EPS);
        ws[b] = -(num / den);
    }
}

// Mean over the 256 per-sample losses.
__global__ void reduce_mean_kernel(const float* __restrict__ ws, float* __restrict__ out) {
    __shared__ float r[256];
    const int t = threadIdx.x;
    r[t] = ws[t];
    __syncthreads();
    #pragma unroll
    for (int s = 128; s > 0; s >>= 1) {
        if (t < s) r[t] += r[t + s];
        __syncthreads();
    }
    if (t == 0) out[0] = r[0] / 256.0f;
}

extern "C" void kernel_launch(void* const* d_in, const int* in_sizes, int n_in,
                              void* d_out, int out_size, void* d_ws, size_t ws_size,
                              hipStream_t stream) {
    const float* pred = (const float*)d_in[0];   // pred_cost [B, D] f32
    const float* ctr  = (const float*)d_in[1];   // tight_ctrs [B, K, D] f32
    float* losses = (float*)d_ws;
    float* out    = (float*)d_out;

    (void)hipFuncSetAttribute((const void*)cone_loss_kernel,
                              hipFuncAttributeMaxDynamicSharedMemorySize,
                              (int)LDS_BYTES);

    cone_loss_kernel<<<dim3(B_SZ), dim3(1024), LDS_BYTES, stream>>>(pred, ctr, losses);
    reduce_mean_kernel<<<dim3(1), dim3(256), 0, stream>>>(losses, out);
}